// ReconstructionLoss_65747359367629
// MI455X (gfx1250) — compile-verified
//
#include <hip/hip_runtime.h>
#include <hip/hip_bf16.h>

// Problem constants from the reference
#define BB   8        // batch
#define PP   64       // basis count (params have P+1 = 65 entries per batch)
#define NPTS 4096     // points per shape
#define CORD 12288    // 3*NPTS coordinates per shape

typedef __attribute__((ext_vector_type(2))) float v2f;
typedef __attribute__((ext_vector_type(4))) float v4f;
typedef __attribute__((ext_vector_type(8))) float v8f;

// ---------------------------------------------------------------------------
// Zero the scalar output (graph-capture safe; harness poisons d_out).
// ---------------------------------------------------------------------------
__global__ void zero_out_kernel(float* __restrict__ out, int n) {
    int i = blockIdx.x * blockDim.x + threadIdx.x;
    if (i < n) out[i] = 0.0f;
}

// ---------------------------------------------------------------------------
// Deform: pts[b][j] = s_b * (mean[j] + sum_k dp[b][k] * basis[k][j])
// params: (B, P+1); mean: (3N); basis: (P, 3N); outPts: (B, 3N)
// ---------------------------------------------------------------------------
__global__ __launch_bounds__(256) void deform_kernel(
    const float* __restrict__ params,
    const float* __restrict__ mean,
    const float* __restrict__ basis,
    float* __restrict__ outPts) {
    const int j = blockIdx.x * blockDim.x + threadIdx.x;
    const int b = blockIdx.y;
    if (j >= CORD) return;
    const float* p = params + b * (PP + 1);
    const float s = p[PP];
    float acc = mean[j];
#pragma unroll 8
    for (int k = 0; k < PP; ++k) {
        acc = fmaf(p[k], basis[k * CORD + j], acc);
    }
    outPts[b * CORD + j] = s * acc;
}

// ---------------------------------------------------------------------------
// Pack each point into a WMMA-ready float4: (p0, p1, p2, |p|^2)
// ---------------------------------------------------------------------------
__global__ __launch_bounds__(256) void pack_kernel(
    const float* __restrict__ pts,   // (B, NPTS, 3)
    v4f* __restrict__ pk) {          // (B*NPTS) float4
    const int i = blockIdx.x * blockDim.x + threadIdx.x;
    if (i >= BB * NPTS) return;
    const float c0 = pts[i * 3 + 0];
    const float c1 = pts[i * 3 + 1];
    const float c2 = pts[i * 3 + 2];
    v4f v;
    v.x = c0; v.y = c1; v.z = c2;
    v.w = c0 * c0 + c1 * c1 + c2 * c2;
    pk[i] = v;
}

// ---------------------------------------------------------------------------
// One-directional chamfer term with V_WMMA_F32_16X16X4_F32.
//
// K=4 folding (norm carried by B):
//   A row M (K0..K3) = (-2*x0, -2*x1, -2*x2, 1.0)
//   B col N (K0..K3) = ( y0,    y1,    y2,  |y_N|^2)
//   => WMMA element (M,N) = |y_N|^2 - 2 * x_M . y_N
// Epilogue: d2 = max(c[i] + |x_M|^2, 0); rmin = min(rmin, d2).
//
// B operand per lane = one aligned b64 load from the packed float4 array:
//   lanes 0-15 (K0,K1) = pk[col].xy ; lanes 16-31 (K2,K3) = pk[col].zw
// -> no selects, no divergence, 1 VMEM per tile.
//
// C/D layout (32-bit 16x16): lane holds 8 elems, (M = i + 8*(lane>=16),
// N = lane&15).  A (16x4 f32): lanes 0-15 -> K=0,1; lanes 16-31 -> K=2,3.
// ---------------------------------------------------------------------------
__global__ __launch_bounds__(128) void chamfer_min_kernel(
    const v4f* __restrict__ Xpk,   // (B*NPTS) packed row points
    const v4f* __restrict__ Ypk,   // (B*NPTS) packed points minimized over
    float* __restrict__ out,
    float scale) {
    const int b    = blockIdx.y;
    const int lane = threadIdx.x & 31;
    const int wave = threadIdx.x >> 5;
    const int nBase = (blockIdx.x * 4 + wave) * 16;

    const v4f* __restrict__ Xb = Xpk + b * NPTS;
    const v4f* __restrict__ Yb = Ypk + b * NPTS;

    const int l16 = lane & 15;
    const bool hi = (lane >= 16);

    // --- A operand: rows nBase..nBase+15
    //     lanes 0-15 : (K0,K1) = (-2*x0, -2*x1)
    //     lanes 16-31: (K2,K3) = (-2*x2, 1.0)
    const v4f xa = Xb[nBase + l16];
    v2f a;
    a.x = hi ? (-2.0f * xa.z) : (-2.0f * xa.x);
    a.y = hi ? 1.0f           : (-2.0f * xa.y);

    // --- |x|^2 for the 8 rows this lane's accumulator slots cover
    float x2[8];
    const int mOff = hi ? 8 : 0;
#pragma unroll
    for (int i = 0; i < 8; ++i) {
        x2[i] = Xb[nBase + mOff + i].w;
    }

    float rmin[8];
#pragma unroll
    for (int i = 0; i < 8; ++i) rmin[i] = 3.402823466e+38f;

    // --- per-lane B pointer: half of packed float4 for column (l16 + m)
    //     (8-byte aligned: base is 16B aligned, hi adds 8 bytes)
    const v2f* __restrict__ Bv = (const v2f*)Yb + (l16 * 2 + (hi ? 1 : 0));

    // --- sweep all Y points in 16-column tiles
#pragma unroll 4
    for (int m = 0; m < NPTS; m += 16) {
        const v2f bv = Bv[m * 2];   // pk[m + l16].xy (lo) or .zw (hi)

        v8f c = {};
        // c(M,N) = |y_N|^2 - 2 x_M . y_N
        c = __builtin_amdgcn_wmma_f32_16x16x4_f32(
                /*neg_a=*/false, a, /*neg_b=*/false, bv,
                /*c_mod=*/(short)0, c, /*reuse_a=*/false, /*reuse_b=*/false);

#pragma unroll
        for (int i = 0; i < 8; ++i) {
            const float d2 = fmaxf(c[i] + x2[i], 0.0f);
            rmin[i] = fminf(rmin[i], d2);
        }
    }

    // --- cross-lane min within each 16-lane column group (wave32)
#pragma unroll
    for (int mask = 8; mask >= 1; mask >>= 1) {
#pragma unroll
        for (int i = 0; i < 8; ++i) {
            rmin[i] = fminf(rmin[i], __shfl_xor(rmin[i], mask, 32));
        }
    }

    // lane 0 holds row mins M=0..7, lane 16 holds M=8..15
    if (l16 == 0) {
        float s = 0.0f;
#pragma unroll
        for (int i = 0; i < 8; ++i) s += rmin[i];
        atomicAdd(out, s * scale);
    }
}

// ---------------------------------------------------------------------------
extern "C" void kernel_launch(void* const* d_in, const int* in_sizes, int n_in,
                              void* d_out, int out_size, void* d_ws, size_t ws_size,
                              hipStream_t stream) {
    const float* pred       = (const float*)d_in[0]; // (8, 65)
    const float* gt         = (const float*)d_in[1]; // (8, 65)
    const float* mean_pred  = (const float*)d_in[2]; // (12288)
    const float* basis_pred = (const float*)d_in[3]; // (64, 12288)
    const float* mean_gt    = (const float*)d_in[4]; // (12288)
    const float* basis_gt   = (const float*)d_in[5]; // (64, 12288)

    float* out = (float*)d_out;
    float* ws  = (float*)d_ws;
    float* predShape = ws;                          // B*CORD floats
    float* gtShape   = predShape + BB * CORD;       // B*CORD floats
    v4f*   predPk    = (v4f*)(gtShape + BB * CORD); // B*NPTS float4 (16B aligned)
    v4f*   gtPk      = predPk + BB * NPTS;          // B*NPTS float4

    // 1) zero the scalar output
    zero_out_kernel<<<dim3((out_size + 255) / 256), dim3(256), 0, stream>>>(out, out_size);

    // 2) deform both shapes into workspace
    {
        dim3 grid((CORD + 255) / 256, BB);
        deform_kernel<<<grid, dim3(256), 0, stream>>>(pred, mean_pred, basis_pred, predShape);
        deform_kernel<<<grid, dim3(256), 0, stream>>>(gt,   mean_gt,   basis_gt,   gtShape);
    }

    // 3) pack points as (x, y, z, |p|^2)
    {
        dim3 grid((BB * NPTS + 255) / 256);
        pack_kernel<<<grid, dim3(256), 0, stream>>>(predShape, predPk);
        pack_kernel<<<grid, dim3(256), 0, stream>>>(gtShape,   gtPk);
    }

    // 4) chamfer: both directions accumulate into out[0]
    //    final = (1/B) * [ mean_n(min_m d2) + mean_m(min_n d2) ]
    {
        const float scale = 1.0f / (float)(BB * NPTS);
        dim3 grid(NPTS / (16 * 4), BB);   // 4 waves/block, 16 rows/wave
        // cham_x: rows = gt points, min over pred points
        chamfer_min_kernel<<<grid, dim3(128), 0, stream>>>(gtPk, predPk, out, scale);
        // cham_y: rows = pred points, min over gt points
        chamfer_min_kernel<<<grid, dim3(128), 0, stream>>>(predPk, gtPk, out, scale);
    }
}